// TMScore_43121471652207
// MI455X (gfx1250) — compile-verified
//
#include <hip/hip_runtime.h>
#include <hip/hip_bf16.h>
#include <math.h>

typedef float v2f __attribute__((ext_vector_type(2)));
typedef float v8f __attribute__((ext_vector_type(8)));

#define TILE 128   // residues per LDS tile in pass 1

// ---------------------------------------------------------------------------
// Kernel 0: mask/true reductions -> ws[0]=Lt, ws[1]=d0^2, ws[2..4]=true_mean
// ---------------------------------------------------------------------------
__global__ __launch_bounds__(256) void tm_setup_kernel(
    const float* __restrict__ truec, const unsigned char* __restrict__ mask,
    float* __restrict__ ws, int L)
{
  __shared__ float r0[256], r1[256], r2[256], r3[256];
  const int tid = threadIdx.x;
  float lt = 0.f, sx = 0.f, sy = 0.f, sz = 0.f;
  for (int l = tid; l < L; l += 256) {
    float w = mask[l] ? 1.f : 0.f;
    lt += w;
    sx += w * truec[3 * l + 0];
    sy += w * truec[3 * l + 1];
    sz += w * truec[3 * l + 2];
  }
  r0[tid] = lt; r1[tid] = sx; r2[tid] = sy; r3[tid] = sz;
  __syncthreads();
  for (int off = 128; off > 0; off >>= 1) {
    if (tid < off) {
      r0[tid] += r0[tid + off]; r1[tid] += r1[tid + off];
      r2[tid] += r2[tid + off]; r3[tid] += r3[tid + off];
    }
    __syncthreads();
  }
  if (tid == 0) {
    float Lt = r0[0];
    float d0 = (Lt <= 15.f) ? 0.5f : (1.24f * cbrtf(Lt - 15.f) - 1.8f);
    d0 = fmaxf(d0, 0.5f);
    ws[0] = Lt;
    ws[1] = d0 * d0;
    ws[2] = r1[0] / Lt;
    ws[3] = r2[0] / Lt;
    ws[4] = r3[0] / Lt;
  }
}

// ---------------------------------------------------------------------------
// Kernel 1 (hot): per block = 4 samples. Computes, via V_WMMA_F32_16X16X4_F32,
//   D[4a+i][n<3]  = H_a[i][n] = sum_l w_l * p_a,l[i] * tc_l[n]
//   D[4a+i][3]    = sum_l w_l * p_a,l[i]          (weighted pred sum)
// A rows (M) = 4 samples x 3 coords, K = 4 residues per wmma, B = [w*tc | w].
// Pred tiles staged to LDS with global_load_async_to_lds_b128 (ASYNCcnt).
// ---------------------------------------------------------------------------
__global__ __launch_bounds__(256) void tm_pass1_kernel(
    const float* __restrict__ pred, const float* __restrict__ truec,
    const unsigned char* __restrict__ mask, const float* __restrict__ ws,
    float* __restrict__ Hout, int L)
{
  __shared__ alignas(16) float pbuf[4][TILE * 3 + 4];  // +4 pad: bank spread
  __shared__ float tbuf[TILE][4];
  __shared__ float red[8][4][3][4];

  const int tid  = threadIdx.x;
  const int lane = tid & 31;
  const int wave = tid >> 5;
  const int h    = lane >> 4;   // lane half (K split)
  const int m    = lane & 15;   // A row / B column index
  const int s0   = blockIdx.x * 4;
  const int samp = m >> 2;      // sample within the 4-sample batch (A rows)
  const int i    = m & 3;       // coordinate row (i==3 -> zero row)

  const float tmx = ws[2], tmy = ws[3], tmz = ws[4];

  v8f acc = {0.f, 0.f, 0.f, 0.f, 0.f, 0.f, 0.f, 0.f};

  const int ntiles = L / TILE;
  for (int tile = 0; tile < ntiles; ++tile) {
    // ---- async copy pred tile: 4 samples x TILE*3 f32 = 384 x 16B chunks ----
    {
      int j = tid;
      #pragma unroll
      for (int it = 0; it < 2; ++it, j += 256) {
        if (j < 4 * (TILE * 3 / 4)) {        // 384 16B chunks
          int sm    = j / (TILE * 3 / 4);    // /96
          int off16 = j % (TILE * 3 / 4);
          const float* g = pred + (size_t)(s0 + sm) * (3 * L)
                                + (size_t)tile * (TILE * 3) + off16 * 4;
          unsigned lds = (unsigned)(size_t)(&pbuf[sm][off16 * 4]);
          asm volatile("global_load_async_to_lds_b128 %0, %1, off"
                       :: "v"(lds), "v"(g) : "memory");
        }
      }
    }
    // ---- B data: tbuf[l][0..2] = w*(true-tmean), tbuf[l][3] = w ----
    for (int j = tid; j < TILE; j += 256) {
      int l = tile * TILE + j;
      float w = mask[l] ? 1.f : 0.f;
      tbuf[j][0] = w * (truec[3 * l + 0] - tmx);
      tbuf[j][1] = w * (truec[3 * l + 1] - tmy);
      tbuf[j][2] = w * (truec[3 * l + 2] - tmz);
      tbuf[j][3] = w;
    }
    asm volatile("s_wait_asynccnt 0" ::: "memory");
    __syncthreads();

    // ---- each wave: 4 wmma chunks (16 residues) of this tile ----
    #pragma unroll
    for (int c = 0; c < 4; ++c) {
      const int r0 = (wave * 4 + c) * 4;  // residue base within tile (K chunk)
      // A 16x4 f32 layout: VGPR0 = K=0 (lanes<16) / K=2 (lanes>=16); VGPR1 = K=1/K=3
      float a0 = 0.f, a1 = 0.f, b0 = 0.f, b1 = 0.f;
      if (i < 3) {
        a0 = pbuf[samp][(r0 + 2 * h + 0) * 3 + i];
        a1 = pbuf[samp][(r0 + 2 * h + 1) * 3 + i];
      }
      // B 4x16 f32 layout: VGPR0 = K=0/K=2 halves, VGPR1 = K=1/K=3; N = lane&15
      if (m < 4) {
        b0 = tbuf[r0 + 2 * h + 0][m];
        b1 = tbuf[r0 + 2 * h + 1][m];
      }
      v2f av = {a0, a1};
      v2f bv = {b0, b1};
      acc = __builtin_amdgcn_wmma_f32_16x16x4_f32(
          false, av, false, bv, (short)0, acc, false, false);
    }
    __syncthreads();
  }

  // ---- extract D: VGPR r holds rows r (lanes0-15) and r+8 (lanes16-31) ----
  if (m < 4) {
    red[wave][0 + 2 * h][0][m] = acc[0];  // rows 0 / 8
    red[wave][0 + 2 * h][1][m] = acc[1];  // rows 1 / 9
    red[wave][0 + 2 * h][2][m] = acc[2];  // rows 2 / 10
    red[wave][1 + 2 * h][0][m] = acc[4];  // rows 4 / 12
    red[wave][1 + 2 * h][1][m] = acc[5];  // rows 5 / 13
    red[wave][1 + 2 * h][2][m] = acc[6];  // rows 6 / 14
  }
  __syncthreads();
  if (tid < 48) {
    int sm = tid / 12, rem = tid % 12, ii = rem >> 2, n = rem & 3;
    float sum = 0.f;
    #pragma unroll
    for (int w = 0; w < 8; ++w) sum += red[w][sm][ii][n];
    Hout[(size_t)(s0 + sm) * 16 + ii * 4 + n] = sum;
  }
}

// ---------------------------------------------------------------------------
// Kernel 2: per-sample proper rotation via Horn's quaternion (f64 power iter).
// Equivalent to reference's  R = V diag(1,1,sign det(VU^T)) U^T.
// Stores Rws[s*12 + 0..8] = R row-major, [9..11] = b = tmean - R*pmean.
// ---------------------------------------------------------------------------
__global__ __launch_bounds__(256) void tm_rot_kernel(
    const float* __restrict__ ws, const float* __restrict__ Hws,
    float* __restrict__ Rws, int S)
{
  int s = blockIdx.x * 256 + threadIdx.x;
  if (s >= S) return;
  const float* Hp = Hws + (size_t)s * 16;
  const double Lt = (double)ws[0];
  double pm[3], H[3][3];
  #pragma unroll
  for (int a = 0; a < 3; ++a) {
    pm[a] = (double)Hp[a * 4 + 3] / Lt;
    #pragma unroll
    for (int b = 0; b < 3; ++b) H[a][b] = (double)Hp[a * 4 + b];
  }
  const double Sxx=H[0][0],Sxy=H[0][1],Sxz=H[0][2];
  const double Syx=H[1][0],Syy=H[1][1],Syz=H[1][2];
  const double Szx=H[2][0],Szy=H[2][1],Szz=H[2][2];
  double K[4][4] = {
    { Sxx+Syy+Szz, Syz-Szy,      Szx-Sxz,      Sxy-Syx      },
    { Syz-Szy,     Sxx-Syy-Szz,  Sxy+Syx,      Szx+Sxz      },
    { Szx-Sxz,     Sxy+Syx,     -Sxx+Syy-Szz,  Syz+Szy      },
    { Sxy-Syx,     Szx+Sxz,      Syz+Szy,     -Sxx-Syy+Szz  }};
  double fro = 0.0;
  #pragma unroll
  for (int a = 0; a < 4; ++a)
    #pragma unroll
    for (int b = 0; b < 4; ++b) fro += K[a][b] * K[a][b];
  const double sig = sqrt(fro) + 1e-300;

  double q[4] = {1.0, 0.0173, -0.0317, 0.0457};  // generic start
  for (int it = 0; it < 200; ++it) {
    double w[4];
    #pragma unroll
    for (int a = 0; a < 4; ++a)
      w[a] = sig * q[a] + K[a][0]*q[0] + K[a][1]*q[1] + K[a][2]*q[2] + K[a][3]*q[3];
    double nrm = sqrt(w[0]*w[0] + w[1]*w[1] + w[2]*w[2] + w[3]*w[3]) + 1e-300;
    #pragma unroll
    for (int a = 0; a < 4; ++a) q[a] = w[a] / nrm;
  }
  const double q0=q[0], q1=q[1], q2=q[2], q3=q[3];
  double R[3][3];
  R[0][0]=q0*q0+q1*q1-q2*q2-q3*q3; R[0][1]=2.0*(q1*q2-q0*q3); R[0][2]=2.0*(q1*q3+q0*q2);
  R[1][0]=2.0*(q1*q2+q0*q3); R[1][1]=q0*q0-q1*q1+q2*q2-q3*q3; R[1][2]=2.0*(q2*q3-q0*q1);
  R[2][0]=2.0*(q1*q3-q0*q2); R[2][1]=2.0*(q2*q3+q0*q1); R[2][2]=q0*q0-q1*q1-q2*q2+q3*q3;

  float* out = Rws + (size_t)s * 12;
  #pragma unroll
  for (int a = 0; a < 3; ++a) {
    #pragma unroll
    for (int b = 0; b < 3; ++b) out[a * 3 + b] = (float)R[a][b];
    double bb = (double)ws[2 + a] - (R[a][0]*pm[0] + R[a][1]*pm[1] + R[a][2]*pm[2]);
    out[9 + a] = (float)bb;
  }
}

// ---------------------------------------------------------------------------
// Kernel 3 (hot): per block = 1 sample. d = R*p + b - t; tm = sum wn/(1+d2/d02)
// pred is L2-resident after pass 1 (50 MB << 192 MB L2).
// ---------------------------------------------------------------------------
__global__ __launch_bounds__(256) void tm_score_kernel(
    const float* __restrict__ pred, const float* __restrict__ truec,
    const unsigned char* __restrict__ mask, const float* __restrict__ ws,
    const float* __restrict__ Rws, float* __restrict__ out, int L)
{
  __shared__ float red[256];
  const int tid = threadIdx.x;
  const int s = blockIdx.x;
  const float* Rb = Rws + (size_t)s * 12;
  const float R00=Rb[0],R01=Rb[1],R02=Rb[2];
  const float R10=Rb[3],R11=Rb[4],R12=Rb[5];
  const float R20=Rb[6],R21=Rb[7],R22=Rb[8];
  const float b0=Rb[9], b1=Rb[10], b2=Rb[11];
  const float Lt = ws[0];
  const float inv_d0sq = 1.f / ws[1];

  const float* P = pred + (size_t)s * 3 * L;
  float acc = 0.f;
  for (int l = tid; l < L; l += 256) {
    float px = P[3*l+0], py = P[3*l+1], pz = P[3*l+2];
    float tx = truec[3*l+0], ty = truec[3*l+1], tz = truec[3*l+2];
    float ax = fmaf(R00,px, fmaf(R01,py, fmaf(R02,pz, b0))) - tx;
    float ay = fmaf(R10,px, fmaf(R11,py, fmaf(R12,pz, b1))) - ty;
    float az = fmaf(R20,px, fmaf(R21,py, fmaf(R22,pz, b2))) - tz;
    float dsq = fmaf(ax,ax, fmaf(ay,ay, az*az));
    float w = mask[l] ? 1.f : 0.f;
    acc += w / fmaf(dsq, inv_d0sq, 1.f);
  }
  red[tid] = acc;
  __syncthreads();
  for (int off = 128; off > 0; off >>= 1) {
    if (tid < off) red[tid] += red[tid + off];
    __syncthreads();
  }
  if (tid == 0) out[s] = red[0] / Lt;
}

// ---------------------------------------------------------------------------
extern "C" void kernel_launch(void* const* d_in, const int* in_sizes, int n_in,
                              void* d_out, int out_size, void* d_ws, size_t ws_size,
                              hipStream_t stream) {
  const float*         pred  = (const float*)d_in[0];          // [S,L,3] f32
  const float*         truec = (const float*)d_in[1];          // [L,3]   f32
  const unsigned char* mask  = (const unsigned char*)d_in[2];  // [L] bool (1B)
  float* outp = (float*)d_out;                                 // [S] f32

  const int L = in_sizes[2];                    // 4096
  const int S = (int)(in_sizes[0] / (3 * L));   // 1024

  float* wsf = (float*)d_ws;
  float* Hws = wsf + 8;                         // S*16 floats
  float* Rws = Hws + (size_t)S * 16;            // S*12 floats

  tm_setup_kernel<<<1, 256, 0, stream>>>(truec, mask, wsf, L);
  tm_pass1_kernel<<<S / 4, 256, 0, stream>>>(pred, truec, mask, wsf, Hws, L);
  tm_rot_kernel<<<(S + 255) / 256, 256, 0, stream>>>(wsf, Hws, Rws, S);
  tm_score_kernel<<<S, 256, 0, stream>>>(pred, truec, mask, wsf, Rws, outp, L);
}